// Clusformer_32976758898708
// MI455X (gfx1250) — compile-verified
//
#include <hip/hip_runtime.h>
#include <math.h>

// Problem constants (match reference)
#define BB  4
#define TT  12
#define NN  4096
#define CC  128
#define LL  32
#define DD  64
#define HH  4
#define HDD 16

typedef float v2f __attribute__((ext_vector_type(2)));
typedef float v8f __attribute__((ext_vector_type(8)));

static __device__ __forceinline__ v8f wmma4(v2f a, v2f b, v8f c) {
  // D = A(16x4,f32) * B(4x16,f32) + C(16x16,f32)
  return __builtin_amdgcn_wmma_f32_16x16x4_f32(false, a, false, b, (short)0, c,
                                               false, false);
}

// ---------------------------------------------------------------------------
// Qc[b,l,c] = centroids[b,l,:] @ Wq_c + bq_c          (tiny: 16384 outputs)
// ---------------------------------------------------------------------------
__global__ void k_qc(const float* __restrict__ cent, const float* __restrict__ Wqc,
                     const float* __restrict__ bqc, float* __restrict__ Qc) {
  int idx = blockIdx.x * blockDim.x + threadIdx.x;
  if (idx >= BB * LL * CC) return;
  int c = idx % CC;
  int bl = idx / CC;
  const float* row = cent + bl * DD;
  float s = bqc[c];
#pragma unroll 8
  for (int d = 0; d < DD; ++d) s += row[d] * Wqc[d * CC + c];
  Qc[idx] = s;
}

// ---------------------------------------------------------------------------
// WeffT[b][l][c] = sum_{c2} Wk_n[c][c2] * Qc[b,l,c2]   (scores folding)
// ---------------------------------------------------------------------------
__global__ void k_weff(const float* __restrict__ Wkn, const float* __restrict__ Qc,
                       float* __restrict__ WeT) {
  int idx = blockIdx.x * blockDim.x + threadIdx.x;
  if (idx >= BB * LL * CC) return;
  int c = idx % CC;
  int bl = idx / CC;
  const float* wrow = Wkn + c * CC;
  const float* qrow = Qc + bl * CC;
  float s = 0.f;
#pragma unroll 8
  for (int c2 = 0; c2 < CC; ++c2) s += wrow[c2] * qrow[c2];
  WeT[idx] = s;
}

// ---------------------------------------------------------------------------
// WvT[n][k] = Wv_n[k][n]; zero cluster/counts; s0[b,l] = bk_n . Qc[b,l,:]
// ---------------------------------------------------------------------------
__global__ void k_prep(const float* __restrict__ Wvn, const float* __restrict__ bkn,
                       const float* __restrict__ Qc, float* __restrict__ WvT,
                       float* __restrict__ cluster, float* __restrict__ counts,
                       float* __restrict__ s0) {
  int idx = blockIdx.x * blockDim.x + threadIdx.x;
  if (idx < CC * CC) {
    int n = idx / CC, k = idx % CC;
    WvT[idx] = Wvn[k * CC + n];
  }
  if (idx < BB * LL * CC) cluster[idx] = 0.f;
  if (idx < BB * LL) {
    counts[idx] = 0.f;
    const float* qrow = Qc + idx * CC;
    float s = 0.f;
#pragma unroll 8
    for (int c = 0; c < CC; ++c) s += bkn[c] * qrow[c];
    s0[idx] = s;
  }
}

// ---------------------------------------------------------------------------
// Heavy fused kernel: per 16-row tile of X[b]:
//   V = X@Wv + bv (WMMA, 8 col-tiles), S = X@Weff + s0 (WMMA, 2 col-tiles),
//   per-row argmax(S) -> assign, LDS scatter-add of V into cluster buckets,
//   workgroup flush to global with float atomics.
// 3072 tiles/batch, stride = gridDim.x*8 waves -> uniform 4 iters per wave.
// ---------------------------------------------------------------------------
__global__ __launch_bounds__(256) void k_cluster(
    const float* __restrict__ X, const float* __restrict__ WvT,
    const float* __restrict__ bvn, const float* __restrict__ WeT,
    const float* __restrict__ s0, float* __restrict__ cluster,
    float* __restrict__ counts) {
  __shared__ float s_cluster[LL * CC];     // 16 KB bucket accumulator
  __shared__ float s_counts[LL];
  __shared__ float s_scores[8 * 16 * 33];  // per-wave 16x32 score tile (padded)
  __shared__ int   s_assign[8 * 16];

  const int tid = threadIdx.x;
  const int wave = tid >> 5;
  const int lane = tid & 31;
  const int half = lane >> 4;   // which K-pair of the fragment this lane holds
  const int l16 = lane & 15;    // row (A) / col (B,C,D) within the 16-tile
  const int b = blockIdx.y;

  for (int i = tid; i < LL * CC; i += 256) s_cluster[i] = 0.f;
  if (tid < LL) s_counts[tid] = 0.f;
  __syncthreads();

  float bvc[8];
#pragma unroll
  for (int t = 0; t < 8; ++t) bvc[t] = bvn[t * 16 + l16];
  float s0v[2];
#pragma unroll
  for (int t = 0; t < 2; ++t) s0v[t] = s0[b * LL + t * 16 + l16];

  const float* Xb = X + (size_t)b * TT * NN * CC;
  const float* WeTb = WeT + b * LL * CC;
  const int ntiles = (TT * NN) / 16;  // 3072
  const int stride = gridDim.x * 8;

  for (int tile = blockIdx.x * 8 + wave; tile < ntiles; tile += stride) {
    const float* xrow = Xb + (size_t)(tile * 16 + l16) * CC + half * 2;

    v8f accV[8], accS[2];
#pragma unroll
    for (int t = 0; t < 8; ++t) {
      float x = bvc[t];
      v8f z = {x, x, x, x, x, x, x, x};
      accV[t] = z;
    }
#pragma unroll
    for (int t = 0; t < 2; ++t) {
      float x = s0v[t];
      v8f z = {x, x, x, x, x, x, x, x};
      accS[t] = z;
    }

#pragma unroll 4
    for (int kk = 0; kk < CC / 4; ++kk) {
      const int k0 = kk * 4;
      v2f a = *(const v2f*)(xrow + k0);  // A frag: rows from X, b64 per lane
#pragma unroll
      for (int t = 0; t < 8; ++t) {      // V projection col-tiles
        v2f bf = *(const v2f*)(WvT + (t * 16 + l16) * CC + k0 + half * 2);
        accV[t] = wmma4(a, bf, accV[t]);
      }
#pragma unroll
      for (int t = 0; t < 2; ++t) {      // score col-tiles (folded Wk@Qc^T)
        v2f bf = *(const v2f*)(WeTb + (t * 16 + l16) * CC + k0 + half * 2);
        accS[t] = wmma4(a, bf, accS[t]);
      }
    }

    // --- scores -> LDS (C/D layout: vgpr j holds M=j+8*half, N=l16) ---
    float* srow = s_scores + wave * 16 * 33;
#pragma unroll
    for (int t = 0; t < 2; ++t)
#pragma unroll
      for (int j = 0; j < 8; ++j)
        srow[(j + half * 8) * 33 + t * 16 + l16] = accS[t][j];
    asm volatile("s_wait_dscnt 0" ::: "memory");  // wave-local RAW on LDS

    if (lane < 16) {  // one lane per row: argmax over L=32
      const float* r = srow + lane * 33;
      float best = r[0];
      int bi = 0;
#pragma unroll
      for (int l = 1; l < LL; ++l) {
        float v = r[l];
        if (v > best) { best = v; bi = l; }
      }
      s_assign[wave * 16 + lane] = bi;
      atomicAdd(&s_counts[bi], 1.0f);
    }
    asm volatile("s_wait_dscnt 0" ::: "memory");

    // --- scatter V rows into LDS cluster buckets ---
#pragma unroll
    for (int j = 0; j < 8; ++j) {
      const int li = s_assign[wave * 16 + j + half * 8];
      float* dst = s_cluster + li * CC + l16;
#pragma unroll
      for (int t = 0; t < 8; ++t) atomicAdd(dst + t * 16, accV[t][j]);
    }
  }

  __syncthreads();
  float* gc = cluster + b * LL * CC;
  for (int i = tid; i < LL * CC; i += 256) atomicAdd(&gc[i], s_cluster[i]);
  if (tid < LL) atomicAdd(&counts[b * LL + tid], s_counts[tid]);
}

// ---------------------------------------------------------------------------
// cen = centroids + (cluster/(counts^2+1)) @ Wal + bal
// ---------------------------------------------------------------------------
__global__ void k_cen(const float* __restrict__ cluster, const float* __restrict__ counts,
                      const float* __restrict__ cent, const float* __restrict__ Wal,
                      const float* __restrict__ bal, float* __restrict__ cen) {
  int idx = blockIdx.x * blockDim.x + threadIdx.x;
  if (idx >= BB * LL * DD) return;
  int d = idx % DD;
  int bl = idx / DD;
  float cnt = counts[bl];
  float f = 1.0f / (cnt * cnt + 1.0f);
  const float* crow = cluster + bl * CC;
  float s = 0.f;
#pragma unroll 8
  for (int c = 0; c < CC; ++c) s += crow[c] * Wal[c * DD + d];
  cen[idx] = cent[idx] + s * f + bal[d];
}

// ---------------------------------------------------------------------------
// q/k/v projections on [B,L,D]
// ---------------------------------------------------------------------------
__global__ void k_qkv(const float* __restrict__ cen, const float* __restrict__ Wq,
                      const float* __restrict__ bq, const float* __restrict__ Wk,
                      const float* __restrict__ bk, const float* __restrict__ Wv,
                      const float* __restrict__ bv, float* __restrict__ q,
                      float* __restrict__ k, float* __restrict__ v) {
  int idx = blockIdx.x * blockDim.x + threadIdx.x;
  if (idx >= BB * LL * DD) return;
  int d = idx % DD;
  int bl = idx / DD;
  const float* row = cen + bl * DD;
  float sq = bq[d], sk = bk[d], sv = bv[d];
#pragma unroll 8
  for (int e = 0; e < DD; ++e) {
    float x = row[e];
    sq += x * Wq[e * DD + d];
    sk += x * Wk[e * DD + d];
    sv += x * Wv[e * DD + d];
  }
  q[idx] = sq;
  k[idx] = sk;
  v[idx] = sv;
}

// ---------------------------------------------------------------------------
// per-(b,h,l) softmax attention over L=32 tokens
// ---------------------------------------------------------------------------
__global__ void k_attn(const float* __restrict__ q, const float* __restrict__ k,
                       const float* __restrict__ v, float* __restrict__ a) {
  int t = blockIdx.x * blockDim.x + threadIdx.x;
  if (t >= BB * HH * LL) return;
  int l = t % LL;
  int h = (t / LL) % HH;
  int b = t / (LL * HH);
  const float* qp = q + (b * LL + l) * DD + h * HDD;
  float sc[LL];
  float mx = -1e30f;
#pragma unroll
  for (int m = 0; m < LL; ++m) {
    const float* kp = k + (b * LL + m) * DD + h * HDD;
    float s = 0.f;
#pragma unroll
    for (int j = 0; j < HDD; ++j) s += qp[j] * kp[j];
    s *= 0.25f;  // 1/sqrt(HD=16)
    sc[m] = s;
    if (s > mx) mx = s;
  }
  float den = 0.f;
#pragma unroll
  for (int m = 0; m < LL; ++m) {
    sc[m] = expf(sc[m] - mx);
    den += sc[m];
  }
  float inv = 1.0f / den;
#pragma unroll
  for (int j = 0; j < HDD; ++j) {
    float s = 0.f;
#pragma unroll
    for (int m = 0; m < LL; ++m) s += sc[m] * v[(b * LL + m) * DD + h * HDD + j];
    a[(b * LL + l) * DD + h * HDD + j] = s * inv;
  }
}

// ---------------------------------------------------------------------------
// z = cen + a @ Wo + bo
// ---------------------------------------------------------------------------
__global__ void k_z(const float* __restrict__ cen, const float* __restrict__ a,
                    const float* __restrict__ Wo, const float* __restrict__ bo,
                    float* __restrict__ z) {
  int idx = blockIdx.x * blockDim.x + threadIdx.x;
  if (idx >= BB * LL * DD) return;
  int d = idx % DD;
  int bl = idx / DD;
  const float* row = a + bl * DD;
  float s = bo[d];
#pragma unroll 8
  for (int e = 0; e < DD; ++e) s += row[e] * Wo[e * DD + d];
  z[idx] = cen[idx] + s;
}

// ---------------------------------------------------------------------------
// BatchNorm stats over (B,L) rows (biased var, eps=1e-5)
// ---------------------------------------------------------------------------
__global__ void k_bn(const float* __restrict__ z, float* __restrict__ mu,
                     float* __restrict__ inv) {
  int d = threadIdx.x;
  if (d >= DD) return;
  float s = 0.f, s2 = 0.f;
  for (int r = 0; r < BB * LL; ++r) {
    float x = z[r * DD + d];
    s += x;
    s2 += x * x;
  }
  float m = s * (1.0f / (BB * LL));
  float var = s2 * (1.0f / (BB * LL)) - m * m;
  mu[d] = m;
  inv[d] = rsqrtf(var + 1e-5f);
}

// ---------------------------------------------------------------------------
// zn = alpha * ((z - mu)*inv*gamma + bn_beta) + beta
// ---------------------------------------------------------------------------
__global__ void k_norm(const float* __restrict__ z, const float* __restrict__ mu,
                       const float* __restrict__ inv, const float* __restrict__ g,
                       const float* __restrict__ bb, const float* __restrict__ alpha,
                       const float* __restrict__ beta, float* __restrict__ zn) {
  int idx = blockIdx.x * blockDim.x + threadIdx.x;
  if (idx >= BB * LL * DD) return;
  int d = idx % DD;
  float x = (z[idx] - mu[d]) * inv[d] * g[d] + bb[d];
  zn[idx] = alpha[d] * x + beta[d];
}

// ---------------------------------------------------------------------------
// h1 = relu(zn @ W1 + b1)
// ---------------------------------------------------------------------------
__global__ void k_ffn1(const float* __restrict__ zn, const float* __restrict__ W1,
                       const float* __restrict__ b1, float* __restrict__ h1) {
  int idx = blockIdx.x * blockDim.x + threadIdx.x;
  if (idx >= BB * LL * DD) return;
  int d = idx % DD;
  int bl = idx / DD;
  const float* row = zn + bl * DD;
  float s = b1[d];
#pragma unroll 8
  for (int e = 0; e < DD; ++e) s += row[e] * W1[e * DD + d];
  h1[idx] = s > 0.f ? s : 0.f;
}

// ---------------------------------------------------------------------------
// out = h1 @ W2 + b2
// ---------------------------------------------------------------------------
__global__ void k_out(const float* __restrict__ h1, const float* __restrict__ W2,
                      const float* __restrict__ b2, float* __restrict__ out) {
  int idx = blockIdx.x * blockDim.x + threadIdx.x;
  if (idx >= BB * LL * DD) return;
  int d = idx % DD;
  int bl = idx / DD;
  const float* row = h1 + bl * DD;
  float s = b2[d];
#pragma unroll 8
  for (int e = 0; e < DD; ++e) s += row[e] * W2[e * DD + d];
  out[idx] = s;
}

// ---------------------------------------------------------------------------
extern "C" void kernel_launch(void* const* d_in, const int* in_sizes, int n_in,
                              void* d_out, int out_size, void* d_ws, size_t ws_size,
                              hipStream_t stream) {
  const float* X    = (const float*)d_in[0];
  const float* cent = (const float*)d_in[1];
  const float* Wqc  = (const float*)d_in[2];
  const float* bqc  = (const float*)d_in[3];
  const float* Wkn  = (const float*)d_in[4];
  const float* bkn  = (const float*)d_in[5];
  const float* Wvn  = (const float*)d_in[6];
  const float* bvn  = (const float*)d_in[7];
  const float* Wal  = (const float*)d_in[8];
  const float* bal  = (const float*)d_in[9];
  const float* Wq   = (const float*)d_in[10];
  const float* bq   = (const float*)d_in[11];
  const float* Wk   = (const float*)d_in[12];
  const float* bk   = (const float*)d_in[13];
  const float* Wv   = (const float*)d_in[14];
  const float* bv   = (const float*)d_in[15];
  const float* Wo   = (const float*)d_in[16];
  const float* bo   = (const float*)d_in[17];
  const float* bng  = (const float*)d_in[18];
  const float* bnb  = (const float*)d_in[19];
  const float* alp  = (const float*)d_in[20];
  const float* bet  = (const float*)d_in[21];
  const float* W1   = (const float*)d_in[22];
  const float* b1   = (const float*)d_in[23];
  const float* W2   = (const float*)d_in[24];
  const float* b2   = (const float*)d_in[25];

  float* ws = (float*)d_ws;
  float* Qc      = ws; ws += BB * LL * CC;   // 16384
  float* WeT     = ws; ws += BB * LL * CC;   // 16384
  float* WvT     = ws; ws += CC * CC;        // 16384
  float* s0      = ws; ws += BB * LL;        // 128
  float* cluster = ws; ws += BB * LL * CC;   // 16384
  float* counts  = ws; ws += BB * LL;        // 128
  float* cen     = ws; ws += BB * LL * DD;   // 8192
  float* qb      = ws; ws += BB * LL * DD;
  float* kb      = ws; ws += BB * LL * DD;
  float* vb      = ws; ws += BB * LL * DD;
  float* ab      = ws; ws += BB * LL * DD;
  float* zb      = ws; ws += BB * LL * DD;
  float* znb     = ws; ws += BB * LL * DD;
  float* h1b     = ws; ws += BB * LL * DD;
  float* mu      = ws; ws += DD;
  float* inv     = ws; ws += DD;

  k_qc<<<64, 256, 0, stream>>>(cent, Wqc, bqc, Qc);
  k_weff<<<64, 256, 0, stream>>>(Wkn, Qc, WeT);
  k_prep<<<64, 256, 0, stream>>>(Wvn, bkn, Qc, WvT, cluster, counts, s0);
  k_cluster<<<dim3(96, BB), 256, 0, stream>>>(X, WvT, bvn, WeT, s0, cluster, counts);
  k_cen<<<32, 256, 0, stream>>>(cluster, counts, cent, Wal, bal, cen);
  k_qkv<<<32, 256, 0, stream>>>(cen, Wq, bq, Wk, bk, Wv, bv, qb, kb, vb);
  k_attn<<<2, 256, 0, stream>>>(qb, kb, vb, ab);
  k_z<<<32, 256, 0, stream>>>(cen, ab, Wo, bo, zb);
  k_bn<<<1, 64, 0, stream>>>(zb, mu, inv);
  k_norm<<<32, 256, 0, stream>>>(zb, mu, inv, bng, bnb, alp, bet, znb);
  k_ffn1<<<32, 256, 0, stream>>>(znb, W1, b1, h1b);
  k_out<<<32, 256, 0, stream>>>(h1b, W2, b2, (float*)d_out);
}